// GCN_28956669510280
// MI455X (gfx1250) — compile-verified
//
#include <hip/hip_runtime.h>

#define HDIM 128

typedef __attribute__((ext_vector_type(2))) float v2f;
typedef __attribute__((ext_vector_type(8))) float v8f;

// ---------------- setup kernels ----------------

__global__ void zero_kernel(int* __restrict__ a, int* __restrict__ b, int n) {
    int i = blockIdx.x * blockDim.x + threadIdx.x;
    if (i < n) { a[i] = 0; b[i] = 0; }
}

__global__ void degree_kernel(const int* __restrict__ dst, int* __restrict__ indeg, int e) {
    int i = blockIdx.x * blockDim.x + threadIdx.x;
    if (i < e) atomicAdd(&indeg[dst[i]], 1);
}

__global__ void dinv_kernel(const int* __restrict__ indeg, float* __restrict__ dinv, int n) {
    int i = blockIdx.x * blockDim.x + threadIdx.x;
    if (i < n) dinv[i] = rsqrtf((float)indeg[i] + 1.0f);   // +1 self-loop
}

// single-block exclusive scan of indeg -> offsets[0..n]
__global__ void __launch_bounds__(1024) scan_kernel(const int* __restrict__ indeg,
                                                    int* __restrict__ offsets, int n) {
    __shared__ int ssum[1024];
    int t = threadIdx.x;
    int chunk = (n + 1023) >> 10;
    int begin = t * chunk;
    int end = begin + chunk; if (end > n) end = n;
    int s = 0;
    for (int i = begin; i < end; ++i) s += indeg[i];
    ssum[t] = s;
    __syncthreads();
    for (int off = 1; off < 1024; off <<= 1) {
        int u = (t >= off) ? ssum[t - off] : 0;
        __syncthreads();
        ssum[t] += u;
        __syncthreads();
    }
    int run = ssum[t] - s;                 // exclusive prefix for this chunk
    for (int i = begin; i < end; ++i) { offsets[i] = run; run += indeg[i]; }
    if (t == 1023) offsets[n] = ssum[1023];
}

__global__ void fill_kernel(const int* __restrict__ dst, const int* __restrict__ offsets,
                            int* __restrict__ cursor, int* __restrict__ csr, int e) {
    int i = blockIdx.x * blockDim.x + threadIdx.x;
    if (i >= e) return;
    int d = dst[i];
    int slot = atomicAdd(&cursor[d], 1);
    csr[offsets[d] + slot] = i;            // store edge id (unique key)
}

// sort each bucket by edge id (deterministic order), then replace id -> src node
__global__ void sortconv_kernel(int* __restrict__ csr, const int* __restrict__ offsets,
                                const int* __restrict__ indeg, const int* __restrict__ esrc,
                                int n) {
    int i = blockIdx.x * blockDim.x + threadIdx.x;
    if (i >= n) return;
    int off = offsets[i], cnt = indeg[i];
    for (int a = 1; a < cnt; ++a) {
        int key = csr[off + a];
        int b = a - 1;
        while (b >= 0 && csr[off + b] > key) { csr[off + b + 1] = csr[off + b]; --b; }
        csr[off + b + 1] = key;
    }
    for (int a = 0; a < cnt; ++a) csr[off + a] = esrc[csr[off + a]];
}

__global__ void embed_kernel(const int* __restrict__ z, const float* __restrict__ table,
                             float* __restrict__ X, int n) {
    int i = blockIdx.x * blockDim.x + threadIdx.x;   // over n*32 float4s
    if (i >= n * 32) return;
    int row = i >> 5, c = i & 31;
    ((float4*)X)[i] = ((const float4*)table)[(size_t)z[row] * 32 + c];
}

// ---------------- WMMA GEMM: Hout[n x 128] = X[n x 128] @ W[128 x 128] ----------------
// 8 waves/block; each wave owns TWO 16-row slabs (32 rows): every B fragment
// feeds 2 WMMAs. W staged in LDS in paired-K layout with XOR-16 column swizzle:
//   sW[(k2*128 + (n ^ ((k2&1)<<4)))*2 + j] = W[2*k2 + j][n]
// so a B fragment is one aligned ds_load_b64, bank-conflict-free across both
// half-waves.
__global__ void __launch_bounds__(256) gemm_kernel(const float* __restrict__ X,
                                                   const float* __restrict__ W,
                                                   float* __restrict__ Hout, int n) {
    __shared__ float sW[HDIM * HDIM];
    int t = threadIdx.x;
    // cooperative staging: 8192 K-pairs / 256 threads = 32 each
    for (int idx = t; idx < HDIM * HDIM / 2; idx += 256) {
        int k2 = idx >> 7;             // paired-K row
        int nn = idx & 127;            // column
        v2f p;
        p.x = W[(2 * k2) * HDIM + nn];
        p.y = W[(2 * k2 + 1) * HDIM + nn];
        *(v2f*)&sW[(k2 * HDIM + (nn ^ ((k2 & 1) << 4))) * 2] = p;
    }
    __syncthreads();

    int wave = t >> 5, lane = t & 31;
    int m0 = (blockIdx.x * 8 + wave) * 32;
    if (m0 >= n) return;                       // after barrier: safe
    int m1 = m0 + 16;
    bool act1 = (m1 < n);

    int lm = lane & 15;                        // M (A) / N (B/D) sub-index
    int kh = lane >> 4;                        // 0/1: half-wave K sub-row

    v8f acc0[8] = {};
    v8f acc1[8] = {};
    const float* xrow0 = X + (size_t)(m0 + lm) * HDIM;
    const float* xrow1 = act1 ? (xrow0 + 16 * HDIM) : xrow0;

    for (int k0 = 0; k0 < HDIM; k0 += 4) {
        int k2 = (k0 >> 1) + kh;               // paired-K row for this half-wave
        v2f a0 = *(const v2f*)(xrow0 + k0 + 2 * kh);
        v2f a1 = *(const v2f*)(xrow1 + k0 + 2 * kh);
        const float* wbase = &sW[(k2 * HDIM + (lm ^ ((k2 & 1) << 4))) * 2];
        #pragma unroll
        for (int tt = 0; tt < 8; ++tt) {
            v2f b = *(const v2f*)(wbase + tt * 32);   // +16 cols = +32 floats
            acc0[tt] = __builtin_amdgcn_wmma_f32_16x16x4_f32(
                false, a0, false, b, (short)0, acc0[tt], false, false);
            acc1[tt] = __builtin_amdgcn_wmma_f32_16x16x4_f32(
                false, a1, false, b, (short)0, acc1[tt], false, false);
        }
    }

    // D layout: lane l<16 -> rows +0..7 ; lane l>=16 -> rows +8..15 ; col = n0+lm
    int rb0 = m0 + (kh << 3);
    int rb1 = m1 + (kh << 3);
    #pragma unroll
    for (int tt = 0; tt < 8; ++tt) {
        int n0 = tt * 16;
        #pragma unroll
        for (int j = 0; j < 8; ++j) {
            Hout[(size_t)(rb0 + j) * HDIM + n0 + lm] = acc0[tt][j];
            if (act1) Hout[(size_t)(rb1 + j) * HDIM + n0 + lm] = acc1[tt][j];
        }
    }
}

// ---------------- pull aggregation: one wave per dst node ----------------
__global__ void __launch_bounds__(256) agg_kernel(const float* __restrict__ Hin,
                                                  float* __restrict__ Xout,
                                                  const float* __restrict__ dinv,
                                                  const int* __restrict__ csr,
                                                  const int* __restrict__ offsets,
                                                  const int* __restrict__ indeg,
                                                  const float* __restrict__ bias,
                                                  int relu, int n) {
    int wave = threadIdx.x >> 5, lane = threadIdx.x & 31;
    int node = blockIdx.x * 8 + wave;
    if (node >= n) return;
    float di = dinv[node];
    const float4* h4 = (const float4*)Hin;
    float4 self = h4[(size_t)node * 32 + lane];
    float ax = di * self.x, ay = di * self.y, az = di * self.z, aw = di * self.w;
    int off = offsets[node], cnt = indeg[node];
    for (int i = 0; i < cnt; ++i) {
        int s = csr[off + i];
        float w = dinv[s];
        float4 hv = h4[(size_t)s * 32 + lane];
        ax += w * hv.x; ay += w * hv.y; az += w * hv.z; aw += w * hv.w;
    }
    float4 b4 = ((const float4*)bias)[lane];
    float4 o;
    o.x = ax * di + b4.x; o.y = ay * di + b4.y;
    o.z = az * di + b4.z; o.w = aw * di + b4.w;
    if (relu) {
        o.x = fmaxf(o.x, 0.f); o.y = fmaxf(o.y, 0.f);
        o.z = fmaxf(o.z, 0.f); o.w = fmaxf(o.w, 0.f);
    }
    ((float4*)Xout)[(size_t)node * 32 + lane] = o;
}

// ---------------- center index + MLP head ----------------
__global__ void ci_kernel(const int* __restrict__ batch, int* __restrict__ ci, int n) {
    int i = blockIdx.x * blockDim.x + threadIdx.x;
    if (i >= n) return;
    if (i == 0 || batch[i] != batch[i - 1]) ci[batch[i]] = i;
}

__global__ void __launch_bounds__(256) mlp_kernel(const float* __restrict__ X,
                                                  const int* __restrict__ ci,
                                                  const float* __restrict__ w1,
                                                  const float* __restrict__ b1,
                                                  const float* __restrict__ w2,
                                                  const float* __restrict__ b2,
                                                  float* __restrict__ out, int G) {
    __shared__ float sxs[8][HDIM];
    int wave = threadIdx.x >> 5, lane = threadIdx.x & 31;
    int g = blockIdx.x * 8 + wave;
    bool active = (g < G);
    if (active) {
        int c = ci[g];
        const float4* x4 = (const float4*)X;
        float4 a = x4[(size_t)c * 32 + lane];
        float4 b = x4[(size_t)(c + 1) * 32 + lane];
        float4 xs;
        xs.x = a.x * b.x; xs.y = a.y * b.y; xs.z = a.z * b.z; xs.w = a.w * b.w;
        ((float4*)sxs[wave])[lane] = xs;
    }
    __syncthreads();
    if (!active) return;
    float4 acc = ((const float4*)b1)[lane];          // lane owns cols lane*4..+3
    for (int k = 0; k < HDIM; ++k) {
        float xk = sxs[wave][k];
        float4 wv = ((const float4*)w1)[k * 32 + lane];
        acc.x += xk * wv.x; acc.y += xk * wv.y;
        acc.z += xk * wv.z; acc.w += xk * wv.w;
    }
    acc.x = fmaxf(acc.x, 0.f); acc.y = fmaxf(acc.y, 0.f);
    acc.z = fmaxf(acc.z, 0.f); acc.w = fmaxf(acc.w, 0.f);
    float4 w2v = ((const float4*)w2)[lane];
    float p = acc.x * w2v.x + acc.y * w2v.y + acc.z * w2v.z + acc.w * w2v.w;
    #pragma unroll
    for (int off = 16; off > 0; off >>= 1) p += __shfl_down(p, off, 32);
    if (lane == 0) out[g] = p + b2[0];
}

// ---------------- launch ----------------

extern "C" void kernel_launch(void* const* d_in, const int* in_sizes, int n_in,
                              void* d_out, int out_size, void* d_ws, size_t ws_size,
                              hipStream_t stream) {
    (void)n_in; (void)ws_size;
    const int*   z      = (const int*)d_in[1];
    const int*   ei     = (const int*)d_in[2];
    const int*   batch  = (const int*)d_in[3];
    const float* table  = (const float*)d_in[4];
    const float* conv_w = (const float*)d_in[5];
    const float* conv_b = (const float*)d_in[6];
    const float* mlp_w1 = (const float*)d_in[7];
    const float* mlp_b1 = (const float*)d_in[8];
    const float* mlp_w2 = (const float*)d_in[9];
    const float* mlp_b2 = (const float*)d_in[10];
    float* out = (float*)d_out;

    const int N = in_sizes[1];
    const int E = in_sizes[2] / 2;
    const int G = out_size;
    const int L = in_sizes[5] / (HDIM * HDIM);

    const int* esrc = ei;
    const int* edst = ei + E;

    // workspace carve-up
    char* ws = (char*)d_ws; size_t off = 0;
    auto alloc = [&](size_t bytes) -> void* {
        void* p = ws + off; off += (bytes + 255) & ~(size_t)255; return p;
    };
    float* xbuf    = (float*)alloc((size_t)N * HDIM * 4);
    float* hbuf    = (float*)alloc((size_t)N * HDIM * 4);
    float* dinv    = (float*)alloc((size_t)N * 4);
    int*   indeg   = (int*)alloc((size_t)N * 4);
    int*   cursor  = (int*)alloc((size_t)N * 4);
    int*   offsets = (int*)alloc((size_t)(N + 1) * 4);
    int*   csr     = (int*)alloc((size_t)E * 4);
    int*   ci      = (int*)alloc((size_t)G * 4);

    const int B = 256;
    int gN  = (N + B - 1) / B;
    int gE  = (E + B - 1) / B;

    zero_kernel<<<gN, B, 0, stream>>>(indeg, cursor, N);
    degree_kernel<<<gE, B, 0, stream>>>(edst, indeg, E);
    dinv_kernel<<<gN, B, 0, stream>>>(indeg, dinv, N);
    scan_kernel<<<1, 1024, 0, stream>>>(indeg, offsets, N);
    fill_kernel<<<gE, B, 0, stream>>>(edst, offsets, cursor, csr, E);
    sortconv_kernel<<<gN, B, 0, stream>>>(csr, offsets, indeg, esrc, N);
    embed_kernel<<<(N * 32 + B - 1) / B, B, 0, stream>>>(z, table, xbuf, N);

    int gGemm = (N + 255) / 256;     // 8 waves x 32 rows per block
    int gAgg  = (N + 7) / 8;         // one wave per node
    for (int l = 0; l < L; ++l) {
        gemm_kernel<<<gGemm, B, 0, stream>>>(xbuf, conv_w + (size_t)l * HDIM * HDIM, hbuf, N);
        agg_kernel<<<gAgg, B, 0, stream>>>(hbuf, xbuf, dinv, csr, offsets, indeg,
                                           conv_b + (size_t)l * HDIM, (l < L - 1) ? 1 : 0, N);
    }

    ci_kernel<<<gN, B, 0, stream>>>(batch, ci, N);
    mlp_kernel<<<(G + 7) / 8, B, 0, stream>>>(xbuf, ci, mlp_w1, mlp_b1, mlp_w2, mlp_b2, out, G);
}